// KeyValueMemoryNet_27264452395468
// MI455X (gfx1250) — compile-verified
//
#include <hip/hip_runtime.h>
#include <hip/hip_bf16.h>
#include <math.h>

typedef __attribute__((ext_vector_type(2))) float v2f;
typedef __attribute__((ext_vector_type(8))) float v8f;

#define D 128
#define MAXN 5.0f

__device__ __forceinline__ float waveSum(float v) {
    #pragma unroll
    for (int off = 16; off > 0; off >>= 1) v += __shfl_xor(v, off, 32);
    return v;
}
__device__ __forceinline__ float waveMax(float v) {
    #pragma unroll
    for (int off = 16; off > 0; off >>= 1) v = fmaxf(v, __shfl_xor(v, off, 32));
    return v;
}

// One wave (32 lanes) encodes one group of L tokens: gather rows, renorm to
// max_norm=5, mean over L. Lane holds dims [lane*4 .. lane*4+3] as float4.
__device__ __forceinline__ float4 encode_group(const int* __restrict__ toks,
                                               const float* __restrict__ emb,
                                               int L, int lane) {
    float4 acc = make_float4(0.f, 0.f, 0.f, 0.f);
    for (int i = 0; i < L; ++i) {
        int tok = toks[i];
        float4 r = *reinterpret_cast<const float4*>(emb + (size_t)tok * D + lane * 4);
        float ss = waveSum(r.x * r.x + r.y * r.y + r.z * r.z + r.w * r.w);
        float s = MAXN / fmaxf(sqrtf(ss), MAXN);   // renorm: shrink only if |x|>5
        acc.x += r.x * s; acc.y += r.y * s; acc.z += r.z * s; acc.w += r.w * s;
    }
    float invL = 1.0f / (float)L;
    acc.x *= invL; acc.y *= invL; acc.z *= invL; acc.w *= invL;
    return acc;
}

// Generic encoder: N groups, L tokens each; out[n*outStride + d]. Optionally
// also stores the L2 norm of the encoded vector (needed for cosine sim).
__global__ void encode_kernel(const int* __restrict__ toks,
                              const float* __restrict__ emb,
                              float* __restrict__ out,
                              float* __restrict__ normOut,
                              int N, int L, int outStride) {
    int wave = (int)((blockIdx.x * blockDim.x + threadIdx.x) >> 5);
    int lane = threadIdx.x & 31;
    if (wave >= N) return;
    float4 acc = encode_group(toks + (size_t)wave * L, emb, L, lane);
    *reinterpret_cast<float4*>(out + (size_t)wave * outStride + lane * 4) = acc;
    if (normOut) {
        float ss = waveSum(acc.x * acc.x + acc.y * acc.y + acc.z * acc.z + acc.w * acc.w);
        if (lane == 0) normOut[wave] = sqrtf(ss);
    }
}

// One wave per (b,m): encode key on the fly, cosine similarity vs q[b].
__global__ void sim_kernel(const int* __restrict__ keys,
                           const float* __restrict__ emb,
                           const float* __restrict__ q,
                           const float* __restrict__ qn,
                           float* __restrict__ sim,
                           int NM, int M, int L) {
    int idx = (int)((blockIdx.x * blockDim.x + threadIdx.x) >> 5);
    int lane = threadIdx.x & 31;
    if (idx >= NM) return;
    int b = idx / M;
    float4 k = encode_group(keys + (size_t)idx * L, emb, L, lane);
    float4 qv = *reinterpret_cast<const float4*>(q + (size_t)b * D + lane * 4);
    float dotv = waveSum(k.x * qv.x + k.y * qv.y + k.z * qv.z + k.w * qv.w);
    float ksq  = waveSum(k.x * k.x + k.y * k.y + k.z * k.z + k.w * k.w);
    if (lane == 0) {
        float denom = fmaxf(qn[b] * sqrtf(ksq), 1e-8f);
        sim[idx] = dotv / denom;
    }
}

// In-place softmax over 512 elements per block (256 threads, 2 elems each).
__global__ void softmax512_kernel(float* __restrict__ sim) {
    int b = blockIdx.x, tid = threadIdx.x;
    __shared__ float red[8];
    float e0 = sim[b * 512 + tid];
    float e1 = sim[b * 512 + 256 + tid];
    float mx = waveMax(fmaxf(e0, e1));
    if ((tid & 31) == 0) red[tid >> 5] = mx;
    __syncthreads();
    mx = red[0];
    #pragma unroll
    for (int i = 1; i < 8; ++i) mx = fmaxf(mx, red[i]);
    __syncthreads();
    float x0 = __expf(e0 - mx), x1 = __expf(e1 - mx);
    float s = waveSum(x0 + x1);
    if ((tid & 31) == 0) red[tid >> 5] = s;
    __syncthreads();
    float tot = 0.f;
    #pragma unroll
    for (int i = 0; i < 8; ++i) tot += red[i];
    float inv = 1.0f / tot;
    sim[b * 512 + tid]       = x0 * inv;
    sim[b * 512 + 256 + tid] = x1 * inv;
}

// Block per b: 16 waves; each wave encodes vals[b,m] on the fly and
// accumulates attn[b,m]-weighted sums; combine waves through LDS.
__global__ void reading_kernel(const int* __restrict__ vals,
                               const float* __restrict__ emb,
                               const float* __restrict__ attn,
                               float* __restrict__ reading,
                               int M, int L) {
    int b = blockIdx.x;
    int wv = threadIdx.x >> 5, lane = threadIdx.x & 31;
    __shared__ float lds[16 * D];
    float4 acc = make_float4(0.f, 0.f, 0.f, 0.f);
    for (int m = wv; m < M; m += 16) {
        float4 v = encode_group(vals + ((size_t)b * M + m) * L, emb, L, lane);
        float a = attn[b * M + m];
        acc.x += v.x * a; acc.y += v.y * a; acc.z += v.z * a; acc.w += v.w * a;
    }
    *reinterpret_cast<float4*>(lds + wv * D + lane * 4) = acc;
    __syncthreads();
    if (threadIdx.x < D) {
        float s = 0.f;
        #pragma unroll
        for (int w = 0; w < 16; ++w) s += lds[w * D + threadIdx.x];
        reading[b * D + threadIdx.x] = s;
    }
}

// result[32,128] = reading[32,128] @ W^T  via V_WMMA_F32_16X16X4_F32,
// then broadcast result[b,:] to x_encoded[b,c,:] for all c (coalesced).
// 16 waves: wave (mi,ni) owns the 16x16 tile at (mi*16, ni*16).
// f32 A 16x4 layout: lanes 0-15 -> M=lane, VGPR0=K0,VGPR1=K1; lanes 16-31 -> K2,K3.
// B[k][n] = W[n][k], so the B fragment addresses W row-major just like A.
__global__ void gemm_broadcast_kernel(const float* __restrict__ reading,
                                      const float* __restrict__ W,
                                      float* __restrict__ xout) {
    int tid = threadIdx.x;
    int w = tid >> 5, lane = tid & 31;
    int mi = w >> 3, ni = w & 7;
    int half = lane >> 4, lr = lane & 15;
    const float* arow = reading + (size_t)(mi * 16 + lr) * D;
    const float* brow = W + (size_t)(ni * 16 + lr) * D;
    v8f acc = {0.f, 0.f, 0.f, 0.f, 0.f, 0.f, 0.f, 0.f};
    #pragma unroll 4
    for (int k = 0; k < D; k += 4) {
        v2f a, bf;
        a.x  = arow[k + 2 * half]; a.y  = arow[k + 2 * half + 1];
        bf.x = brow[k + 2 * half]; bf.y = brow[k + 2 * half + 1];
        acc = __builtin_amdgcn_wmma_f32_16x16x4_f32(
            /*neg_a=*/false, a, /*neg_b=*/false, bf,
            /*c_mod=*/(short)0, acc, /*reuse_a=*/false, /*reuse_b=*/false);
    }
    __shared__ float res[32 * D];   // result[b][d]
    #pragma unroll
    for (int g = 0; g < 8; ++g) {
        int row = mi * 16 + half * 8 + g;   // C/D layout: VGPR g = M=g (lanes 0-15), M=8+g (16-31)
        int col = ni * 16 + lr;
        res[row * D + col] = acc[g];
    }
    __syncthreads();
    // x_encoded[b,c,d] = result[b,d]; 32*128*128 elems, coalesced stores.
    for (int idx = tid; idx < 32 * 128 * D; idx += 512) {
        int b = idx >> 14;
        int d = idx & (D - 1);
        xout[idx] = res[b * D + d];
    }
}

extern "C" void kernel_launch(void* const* d_in, const int* in_sizes, int n_in,
                              void* d_out, int out_size, void* d_ws, size_t ws_size,
                              hipStream_t stream) {
    const int B = 32, Lq = 32, Lr = 20, M = 512, Lk = 32, C = 128, Lc = 20;
    const int* query    = (const int*)d_in[0];
    const int* response = (const int*)d_in[1];
    const int* mkeys    = (const int*)d_in[2];
    const int* mvals    = (const int*)d_in[3];
    const int* cands    = (const int*)d_in[4];
    const float* emb_in  = (const float*)d_in[5];
    const float* emb_out = (const float*)d_in[6];
    const float* W       = (const float*)d_in[7];

    float* out = (float*)d_out;
    float* x = out;                          // [B,C,D]
    float* y = out + (size_t)B * C * D;      // [B,C,D]

    float* ws      = (float*)d_ws;
    float* q       = ws;                     // B*D     = 4096
    float* qn      = ws + 4096;              // B       = 32
    float* sim     = ws + 4224;              // B*M     = 16384 (becomes attn in place)
    float* reading = ws + 20608;             // B*D     = 4096

    // Encodes: query -> ws (+norms); candidates -> y; response -> y[:,0,:]
    encode_kernel<<<(B + 7) / 8, 256, 0, stream>>>(query, emb_in, q, qn, B, Lq, D);
    encode_kernel<<<(B * C + 7) / 8, 256, 0, stream>>>(cands, emb_out, y, nullptr, B * C, Lc, D);
    encode_kernel<<<(B + 7) / 8, 256, 0, stream>>>(response, emb_out, y, nullptr, B, Lr, C * D);

    // Cosine similarity, softmax, attention-weighted value read
    sim_kernel<<<(B * M + 7) / 8, 256, 0, stream>>>(mkeys, emb_in, q, qn, sim, B * M, M, Lk);
    softmax512_kernel<<<B, 256, 0, stream>>>(sim);
    reading_kernel<<<B, 512, 0, stream>>>(mvals, emb_in, sim, reading, M, Lk);

    // result = reading @ W^T (WMMA f32) + broadcast to x_encoded
    gemm_broadcast_kernel<<<1, 512, 0, stream>>>(reading, W, x);
}